// FlexibleGNN_83837761618516
// MI455X (gfx1250) — compile-verified
//
#include <hip/hip_runtime.h>

// GCN: N=50000 nodes, E=600000 edges, LOOKBACK=96, HIDDEN=128, HORIZON=24, 3 layers.
#define LOOKBACK 96
#define HIDDEN   128
#define HORIZON  24
#define N_LAYERS 3

typedef __attribute__((ext_vector_type(2))) float v2f;
typedef __attribute__((ext_vector_type(8))) float v8f;

// ---------------------------------------------------------------------------
// fp32 WMMA GEMM: C[M x NV] = A[M x K] @ B[K x NV] (+ bias), K/NV compile-time.
// One wave computes one 16x16 tile of C via V_WMMA_F32_16X16X4_F32.
// M must be a multiple of 16; K a multiple of 4. NV may be ragged (NV=24):
// B loads use a clamped (always-valid) column + branchless select-to-zero, so
// no EXEC manipulation happens in the K-loop and EXEC is all-1s at every WMMA.
// Full unroll => all load addresses are base + constant immediate offsets.
//
// Lane layouts per CDNA5 ISA 7.12.2:
//   A (16x4):  lanes 0-15: row M=lane,    K=k0+{0,1} in v0/v1;
//              lanes 16-31: row M=lane-16, K=k0+{2,3}.
//   B (4x16):  mirrored (K-rows striped across the two lane halves).
//   C (16x16): VGPR v, lanes 0-15 -> M=v, lanes 16-31 -> M=v+8; N=lane%16.
// ---------------------------------------------------------------------------
template <int K, int NV>
__global__ void gcn_wmma_gemm_f32(const float* __restrict__ A,
                                  const float* __restrict__ B,
                                  const float* __restrict__ bias,
                                  float* __restrict__ C, int M) {
    constexpr int TILES_N = (NV + 15) / 16;
    constexpr bool ALIGNED = (NV % 16) == 0;

    const int lane   = threadIdx.x & 31;
    const int wave   = threadIdx.x >> 5;
    const int tile   = blockIdx.x * (blockDim.x >> 5) + wave;
    const int tilesM = M >> 4;
    if (tile >= tilesM * TILES_N) return;  // wave-uniform guard

    const int tm   = tile / TILES_N;
    const int tn   = tile - tm * TILES_N;
    const int half = lane >> 4;            // 0: K pair {0,1}; 1: K pair {2,3}
    const int lm   = lane & 15;

    const int  m   = tm * 16 + lm;         // A row handled by this lane
    const int  n   = tn * 16 + lm;         // B/C column handled by this lane
    const bool nok = ALIGNED || (n < NV);
    const int  nc  = nok ? n : (NV - 1);   // clamped, always-valid B column

    const float* __restrict__ ap = A + (size_t)m * K + half * 2;
    const float* __restrict__ bp = B + (size_t)(half * 2) * NV + nc;

    v8f acc = {0.f, 0.f, 0.f, 0.f, 0.f, 0.f, 0.f, 0.f};

    #pragma unroll
    for (int k0 = 0; k0 < K; k0 += 4) {
        v2f a, b;
        a.x = ap[k0];                      // constant immediate offsets
        a.y = ap[k0 + 1];
        const float b0 = bp[(size_t)k0 * NV];
        const float b1 = bp[(size_t)(k0 + 1) * NV];
        b.x = nok ? b0 : 0.f;              // compile-time dead when ALIGNED
        b.y = nok ? b1 : 0.f;
        // D = A x B + C  (neg_a, A, neg_b, B, c_mod, C, reuse_a, reuse_b)
        acc = __builtin_amdgcn_wmma_f32_16x16x4_f32(
                  false, a, false, b, (short)0, acc, false, false);
    }

    if (nok) {
        const float bv = bias ? bias[n] : 0.0f;
        #pragma unroll
        for (int v = 0; v < 8; ++v) {
            const int mr = tm * 16 + half * 8 + v;
            C[(size_t)mr * NV + n] = acc[v] + bv;
        }
    }
}

// ---------------------------------------------------------------------------
// Utility kernels
// ---------------------------------------------------------------------------
__global__ void zero_f32v4(float4* __restrict__ p, long long n4) {
    long long t = (long long)blockIdx.x * blockDim.x + threadIdx.x;
    if (t < n4) p[t] = make_float4(0.f, 0.f, 0.f, 0.f);
}

// deg[dst[e]] += 1 over real edges (self-loops folded into make_dinv)
__global__ void deg_count(const long long* __restrict__ dst,
                          float* __restrict__ deg, int E) {
    int e = blockIdx.x * blockDim.x + threadIdx.x;
    if (e < E) atomicAdd(&deg[dst[e]], 1.0f);
}

// dinv[i] = rsqrt(deg[i] + 1)   (+1 = self-loop, guarantees deg > 0)
__global__ void make_dinv(float* __restrict__ deg, int n) {
    int i = blockIdx.x * blockDim.x + threadIdx.x;
    if (i < n) deg[i] = rsqrtf(deg[i] + 1.0f);
}

// agg[dst] += dinv[src]*dinv[dst] * hw[src]; one wave per edge, 32 lanes cover
// the 128 features as float4 chunks (coalesced gather; uniform src/dst/dinv
// reads hit one cacheline and broadcast).
__global__ void edge_scatter(const long long* __restrict__ src,
                             const long long* __restrict__ dst,
                             const float* __restrict__ dinv,
                             const float* __restrict__ hw,
                             float* __restrict__ agg, int E) {
    long long t = (long long)blockIdx.x * blockDim.x + threadIdx.x;
    const int e = (int)(t >> 5);
    const int c = (int)(t & 31);
    if (e >= E) return;
    const long long s = src[e];
    const long long d = dst[e];
    const float w = dinv[s] * dinv[d];
    const float4 v = ((const float4*)(hw + s * HIDDEN))[c];
    float* o = agg + d * HIDDEN + c * 4;
    atomicAdd(o + 0, v.x * w);
    atomicAdd(o + 1, v.y * w);
    atomicAdd(o + 2, v.z * w);
    atomicAdd(o + 3, v.w * w);
}

// h = relu(agg + dinv^2 * hw + bias)   (dinv^2*hw = analytic self-loop term)
__global__ void finalize_layer(const float* __restrict__ agg,
                               const float* __restrict__ hw,
                               const float* __restrict__ dinv,
                               const float* __restrict__ bias,
                               float* __restrict__ h, int N) {
    long long t = (long long)blockIdx.x * blockDim.x + threadIdx.x;
    if (t >= (long long)N * HIDDEN) return;
    const int i = (int)(t >> 7);          // node
    const int j = (int)(t & (HIDDEN - 1));
    const float di = dinv[i];
    const float v = agg[t] + di * di * hw[t] + bias[j];
    h[t] = fmaxf(v, 0.0f);
}

// ---------------------------------------------------------------------------
extern "C" void kernel_launch(void* const* d_in, const int* in_sizes, int n_in,
                              void* d_out, int out_size, void* d_ws, size_t ws_size,
                              hipStream_t stream) {
    const float*     x     = (const float*)d_in[0];
    const float*     y     = (const float*)d_in[1];
    const long long* ei    = (const long long*)d_in[2];   // int64 edge_index [2, E]
    const float*     W_in  = (const float*)d_in[3];
    const float*     b_in  = (const float*)d_in[4];
    const float*     Ws    = (const float*)d_in[5];
    const float*     bs    = (const float*)d_in[6];
    const float*     W_out = (const float*)d_in[7];
    const float*     b_out = (const float*)d_in[8];

    const int N = in_sizes[0] / LOOKBACK;   // 50000 (multiple of 16)
    const int E = in_sizes[2] / 2;          // 600000
    const long long* src = ei;
    const long long* dst = ei + E;

    // Workspace layout (floats): dinv[N] | h[N*H] | hw[N*H] | agg[N*H]  (~77MB)
    float* ws   = (float*)d_ws;
    float* dinv = ws;
    float* h    = ws + N;
    float* hw   = h + (size_t)N * HIDDEN;
    float* agg  = hw + (size_t)N * HIDDEN;

    const int T = 256;                       // 8 waves per block (wave32)
    const long long NH = (long long)N * HIDDEN;

    // --- degrees / symmetric normalization ---
    {
        const long long n4 = N / 4;          // 50000 % 4 == 0
        zero_f32v4<<<(int)((n4 + T - 1) / T), T, 0, stream>>>((float4*)dinv, n4);
    }
    deg_count<<<(E + T - 1) / T, T, 0, stream>>>(dst, dinv, E);
    make_dinv<<<(N + T - 1) / T, T, 0, stream>>>(dinv, N);

    const int tilesH = (N / 16) * (HIDDEN / 16);

    // --- input projection: h = x @ W_in + b_in ---
    gcn_wmma_gemm_f32<LOOKBACK, HIDDEN><<<(tilesH + 7) / 8, T, 0, stream>>>(
        x, W_in, b_in, h, N);

    // --- GCN layers ---
    for (int l = 0; l < N_LAYERS; ++l) {
        gcn_wmma_gemm_f32<HIDDEN, HIDDEN><<<(tilesH + 7) / 8, T, 0, stream>>>(
            h, Ws + (size_t)l * HIDDEN * HIDDEN, nullptr, hw, N);

        const long long nh4 = NH / 4;
        zero_f32v4<<<(int)((nh4 + T - 1) / T), T, 0, stream>>>((float4*)agg, nh4);

        const long long et = (long long)E * 32;   // 32 float4 chunks per edge
        edge_scatter<<<(int)((et + T - 1) / T), T, 0, stream>>>(
            src, dst, dinv, hw, agg, E);

        finalize_layer<<<(int)((NH + T - 1) / T), T, 0, stream>>>(
            agg, hw, dinv, bs + (size_t)l * HIDDEN, h, N);
    }

    // --- output projection straight into d_out: out = h @ W_out + b_out ---
    {
        const int tilesO = (N / 16) * ((HORIZON + 15) / 16);  // ragged N-tile
        gcn_wmma_gemm_f32<HIDDEN, HORIZON><<<(tilesO + 7) / 8, T, 0, stream>>>(
            h, W_out, b_out, (float*)d_out, N);
    }

    // --- tuple second element: pass-through copy of y ---
    hipMemcpyAsync((float*)d_out + (size_t)N * HORIZON, y,
                   sizeof(float) * (size_t)N * HORIZON,
                   hipMemcpyDeviceToDevice, stream);
}